// LaplacianLoss_60155311947792
// MI455X (gfx1250) — compile-verified
//
#include <hip/hip_runtime.h>
#include <hip/hip_bf16.h>
#include <stddef.h>

#define NV   12288
#define NB   8
#define NCOL 24
#define LDB  32   // padded columns (24 real + 8 zero) -> two 16-wide WMMA tiles

typedef __attribute__((ext_vector_type(2))) float v2f;
typedef __attribute__((ext_vector_type(8))) float v8f;

// ---------------------------------------------------------------------------
// Kernel 1: pack x[B][NV][3] into zero-padded column-matrix Xrp[NV][32],
// column n = b*3 + d for n < 24, zero otherwise.
// ---------------------------------------------------------------------------
__global__ __launch_bounds__(256) void lap_pack(const float* __restrict__ x,
                                                float* __restrict__ Bp) {
    int idx = blockIdx.x * 256 + threadIdx.x;   // over NV*LDB
    int k = idx >> 5;
    int n = idx & 31;
    float v = 0.0f;
    if (n < NCOL) {
        int b = n / 3;
        int d = n - 3 * b;
        v = x[((size_t)b * NV + (size_t)k) * 3 + d];
    }
    Bp[idx] = v;
}

// ---------------------------------------------------------------------------
// Kernel 2: split-K GEMM  P[seg][NV][32] = L[strip rows] * Xrp  via
// V_WMMA_F32_16X16X4_F32.  One wave32 per (16-row strip, K segment).
//
// A layout (32-bit A 16x4, ISA 7.12.2): lane = h*16 + m (h = lane>>4) holds
// VGPR pair = { L[row m, k+2h], L[row m, k+2h+1] }  -> one b64 load per lane.
// B layout (4x16): VGPR0 = row K=2h across lanes, VGPR1 = row K=2h+1.
// C/D layout: VGPR r, lane h*16+n = C[r + 8h, n].
// ---------------------------------------------------------------------------
__global__ __launch_bounds__(32) void lap_gemm(const float* __restrict__ L,
                                               const float* __restrict__ Bp,
                                               float* __restrict__ P,
                                               int seg_len) {
    const int strip = blockIdx.x;       // 0 .. NV/16-1
    const int seg   = blockIdx.y;       // 0 .. ksplit-1
    const int lane  = threadIdx.x;      // 0 .. 31
    const int m     = lane & 15;
    const int h     = lane >> 4;
    const int k0    = seg * seg_len;

    const float* __restrict__ a  = L  + (size_t)(strip * 16 + m) * NV + (size_t)k0 + 2 * h;
    const float* __restrict__ b0 = Bp + (size_t)(k0 + 2 * h) * LDB + m;       // cols 0..15
    const float* __restrict__ b1 = b0 + 16;                                    // cols 16..31

    v8f acc0 = {0.f, 0.f, 0.f, 0.f, 0.f, 0.f, 0.f, 0.f};
    v8f acc1 = {0.f, 0.f, 0.f, 0.f, 0.f, 0.f, 0.f, 0.f};

#pragma unroll 4
    for (int k = 0; k < seg_len; k += 4) {
        // A tile 16x4: 256B per wave per step, streamed once from HBM
        v2f av = *(const v2f*)a;
        a += 4;
        // B tiles from L2-resident packed Xrp
        v2f bv0, bv1;
        bv0.x = b0[0];
        bv0.y = b0[LDB];
        bv1.x = b1[0];
        bv1.y = b1[LDB];
        b0 += 4 * LDB;
        b1 += 4 * LDB;

        acc0 = __builtin_amdgcn_wmma_f32_16x16x4_f32(false, av, false, bv0,
                                                     (short)0, acc0, false, false);
        acc1 = __builtin_amdgcn_wmma_f32_16x16x4_f32(false, av, false, bv1,
                                                     (short)0, acc1, false, false);
    }

    // write partial tile: P[seg][strip*16 + row][col]
    float* __restrict__ p = P + ((size_t)seg * NV + (size_t)strip * 16) * LDB;
#pragma unroll
    for (int r = 0; r < 8; ++r) {
        int row = r + 8 * h;
        p[(size_t)row * LDB + m]      = acc0[r];
        p[(size_t)row * LDB + m + 16] = acc1[r];
    }
}

// ---------------------------------------------------------------------------
// Kernel 3: out[b][v] = sum_d ( sum_seg P[seg][v][b*3+d] )^2
// ---------------------------------------------------------------------------
__global__ __launch_bounds__(256) void lap_reduce(const float* __restrict__ P,
                                                  float* __restrict__ out,
                                                  int ksplit) {
    int idx = blockIdx.x * 256 + threadIdx.x;   // over NB*NV
    if (idx >= NB * NV) return;
    int b = idx / NV;
    int v = idx - b * NV;
    int n0 = b * 3;
    float s = 0.0f;
#pragma unroll
    for (int d = 0; d < 3; ++d) {
        float acc = 0.0f;
        for (int q = 0; q < ksplit; ++q)
            acc += P[((size_t)q * NV + (size_t)v) * LDB + n0 + d];
        s += acc * acc;
    }
    out[idx] = s;   // out is [B, NV] row-major
}

// ---------------------------------------------------------------------------
extern "C" void kernel_launch(void* const* d_in, const int* in_sizes, int n_in,
                              void* d_out, int out_size, void* d_ws, size_t ws_size,
                              hipStream_t stream) {
    const float* x = (const float*)d_in[0];   // [B, NV, 3]
    const float* L = (const float*)d_in[1];   // [NV, NV]
    float* out = (float*)d_out;               // [B, NV]

    const size_t xrp_elems = (size_t)NV * LDB;                // 1.5 MB
    const size_t part_elems_per_seg = (size_t)NV * LDB;       // 1.5 MB each

    // choose split-K from available workspace (deterministic per launch config)
    int ksplit = 1;
    for (int ks = 8; ks >= 1; ks >>= 1) {
        size_t need = (xrp_elems + part_elems_per_seg * (size_t)ks) * sizeof(float);
        if (need <= ws_size) { ksplit = ks; break; }
    }

    float* xrp     = (float*)d_ws;
    float* partial = xrp + xrp_elems;
    const int seg_len = NV / ksplit;   // NV divisible by 1,2,4,8; stays multiple of 4

    lap_pack<<<(NV * LDB) / 256, 256, 0, stream>>>(x, xrp);

    dim3 g(NV / 16, ksplit);
    lap_gemm<<<g, 32, 0, stream>>>(L, xrp, partial, seg_len);

    lap_reduce<<<(NB * NV + 255) / 256, 256, 0, stream>>>(partial, out, ksplit);
}